// DisplaceChannel_18717467476169
// MI455X (gfx1250) — compile-verified
//
#include <hip/hip_runtime.h>
#include <stdint.h>

// DisplaceChannel: B=16, C=384, H=64, W=64; 48 groups of 8 channels,
// offsets (iw*16, ih*16), (ih,iw) in 7x7 raster minus center.
// out[b,c,y,x] = x[b,c,y-oy,x-ox] if in-bounds else 0.
//
// Grid = (4, C, B): one block = quarter of one (b,c) plane (256 float4s).
// c is block-uniform, so all group/offset math lowers to SALU.
// Valid lanes: 16B routed through the gfx1250 async global<->LDS engine
// (ASYNCcnt) with non-temporal hints (streaming, zero reuse).
// Invalid lanes: direct non-temporal global_store_b128 of zeros.

#define TPB 256

typedef __attribute__((ext_vector_type(4))) float v4f;

__global__ __launch_bounds__(TPB) void displace_async_kernel(
    const float* __restrict__ x, float* __restrict__ out) {
  __shared__ float smem[TPB * 4];  // one float4 staging slot per thread

  // ---- uniform (SALU) group math ----
  const int c   = blockIdx.y;          // [0,384)
  const int b   = blockIdx.z;          // [0,16)
  const int p   = c >> 3;              // group [0,48)
  const int pp  = p + (p >= 24);       // skip center of 7x7 raster
  const int ih  = pp / 7 - 3;          // [-3,3]
  const int iw  = (pp % 7) - 3;        // [-3,3]
  const int oy  = ih * 16;             // row offset
  const int oxv = iw * 4;              // col offset in float4 units

  // ---- per-lane (VALU) ----
  const int t   = blockIdx.x * TPB + threadIdx.x;  // [0,1024) in plane
  const int xv  = t & 15;              // float4 column [0,16)
  const int y   = t >> 4;              // row [0,64)
  const int sy  = y - oy;
  const int sxv = xv - oxv;
  const bool valid = ((unsigned)sy < 64u) && ((unsigned)sxv < 16u);

  const size_t plane4 = ((size_t)(b * 384 + c)) << 10;  // plane base, float4s
  float* dst = out + ((plane4 + (size_t)t) << 2);

  // LDS byte offset = low 32 bits of flat shared address.
  const uint32_t lds_off =
      (uint32_t)(uintptr_t)(&smem[(int)threadIdx.x * 4]);

  if (valid) {
    const float* src = x + ((plane4 + (size_t)(sy * 16 + sxv)) << 2);
    // memory -> LDS, no VGPR data movement (ASYNCcnt), streaming hint
    asm volatile("global_load_async_to_lds_b128 %0, %1, off th:TH_LOAD_NT"
                 :
                 : "v"(lds_off), "v"((uint64_t)(uintptr_t)src)
                 : "memory");
  }

  // Async loads must land in LDS before the async store reads it.
  asm volatile("s_wait_asynccnt 0" ::: "memory");

  if (valid) {
    // LDS -> memory (ASYNCcnt); drained by s_endpgm implicit wait-idle.
    asm volatile("global_store_async_from_lds_b128 %0, %1, off th:TH_STORE_NT"
                 :
                 : "v"((uint64_t)(uintptr_t)dst), "v"(lds_off)
                 : "memory");
  } else {
    v4f z = (v4f)(0.0f);
    __builtin_nontemporal_store(z, (v4f*)dst);  // NT global_store_b128
  }
}

extern "C" void kernel_launch(void* const* d_in, const int* in_sizes, int n_in,
                              void* d_out, int out_size, void* d_ws, size_t ws_size,
                              hipStream_t stream) {
  (void)in_sizes; (void)n_in; (void)d_ws; (void)ws_size; (void)out_size;
  const float* x = (const float*)d_in[0];
  float* out = (float*)d_out;

  // B=16, C=384, H*W/4 = 1024 float4s per plane -> 4 blocks of 256.
  dim3 grid(4, 384, 16);
  displace_async_kernel<<<grid, TPB, 0, stream>>>(x, out);
}